// BilinearInteraction_26456998543821
// MI455X (gfx1250) — compile-verified
//
#include <hip/hip_runtime.h>

typedef float v2f __attribute__((ext_vector_type(2)));
typedef float v8f __attribute__((ext_vector_type(8)));

#define B_TOT 4096
#define F_    32
#define D_    64
#define P_    496
#define BT    128   // batch rows per workgroup
#define WS    68    // padded LDS row stride (floats): 272B, 16B-aligned, rotates banks

__global__ __launch_bounds__(256)
void bilinear_wmma_f32_kernel(const float* __restrict__ inp,
                              const float* __restrict__ W,
                              float* __restrict__ out)
{
    __shared__ float ldsW[D_ * WS];   // 64 x 68 floats = 17.0 KB
    __shared__ float ldsA[BT * WS];   // 128 x 68 floats = 34.0 KB

    const int tid = threadIdx.x;
    const int p   = blockIdx.y;        // pair index 0..495
    const int b0  = blockIdx.x * BT;   // batch tile base

    // pair p -> (i, j), uniform scalar loop (<=31 iters)
    int i = 0, rem = p;
    while (rem >= F_ - 1 - i) { rem -= F_ - 1 - i; ++i; }
    const int j = i + 1 + rem;

    // ---- stage W[p] (64x64 f32) into LDS, float4-coalesced ----
    {
        const float4* Wp = (const float4*)(W + (size_t)p * D_ * D_); // 1024 float4
        #pragma unroll
        for (int q = 0; q < 4; ++q) {
            int id  = tid + q * 256;
            int row = id >> 4, c4 = id & 15;
            *(float4*)&ldsW[row * WS + c4 * 4] = Wp[id];
        }
    }
    // ---- stage vi tile (128x64 f32) into LDS, float4-coalesced ----
    {
        #pragma unroll
        for (int q = 0; q < 8; ++q) {
            int id  = tid + q * 256;   // 2048 float4 total
            int row = id >> 4, c4 = id & 15;
            const float4* src =
                (const float4*)(inp + ((size_t)(b0 + row) * F_ + i) * D_ + c4 * 4);
            *(float4*)&ldsA[row * WS + c4 * 4] = *src;
        }
    }
    __syncthreads();

    const int wave  = tid >> 5;
    const int lane  = tid & 31;
    const int half  = lane >> 4;   // 0: lanes 0-15, 1: lanes 16-31
    const int lm    = lane & 15;
    const int row16 = wave * 16;   // this wave's 16-row M block

    v8f acc0 = {}, acc1 = {}, acc2 = {}, acc3 = {};

    const float* aRow = &ldsA[(row16 + lm) * WS];

    #pragma unroll
    for (int kk = 0; kk < 16; ++kk) {
        const int kb = kk * 4 + half * 2;   // lane's K base within this K-step
        // A fragment: lane holds M=lm, K = kb+0 (vgpr0), kb+1 (vgpr1)
        v2f a;
        a.x = aRow[kb + 0];
        a.y = aRow[kb + 1];
        // B fragments: lane holds N = nt*16+lm, K = kb+0 / kb+1
        const float* w0 = &ldsW[(kb + 0) * WS + lm];
        const float* w1 = &ldsW[(kb + 1) * WS + lm];
        v2f b0v; b0v.x = w0[ 0]; b0v.y = w1[ 0];
        v2f b1v; b1v.x = w0[16]; b1v.y = w1[16];
        v2f b2v; b2v.x = w0[32]; b2v.y = w1[32];
        v2f b3v; b3v.x = w0[48]; b3v.y = w1[48];
        acc0 = __builtin_amdgcn_wmma_f32_16x16x4_f32(false, a, false, b0v, (short)0, acc0, false, false);
        acc1 = __builtin_amdgcn_wmma_f32_16x16x4_f32(false, a, false, b1v, (short)0, acc1, false, false);
        acc2 = __builtin_amdgcn_wmma_f32_16x16x4_f32(false, a, false, b2v, (short)0, acc2, false, false);
        acc3 = __builtin_amdgcn_wmma_f32_16x16x4_f32(false, a, false, b3v, (short)0, acc3, false, false);
    }

    // ---- epilogue: multiply by vj and store ----
    // C/D layout: VGPR v holds M = v + 8*half, N = lm  (per 16x16 f32 table)
    v8f accs[4] = {acc0, acc1, acc2, acc3};
    #pragma unroll
    for (int nt = 0; nt < 4; ++nt) {
        const int gcol = nt * 16 + lm;
        #pragma unroll
        for (int v = 0; v < 8; ++v) {
            const int grow = b0 + row16 + half * 8 + v;
            const float vj = inp[((size_t)grow * F_ + j) * D_ + gcol];
            out[((size_t)grow * P_ + p) * D_ + gcol] = accs[nt][v] * vj;
        }
    }
}

extern "C" void kernel_launch(void* const* d_in, const int* in_sizes, int n_in,
                              void* d_out, int out_size, void* d_ws, size_t ws_size,
                              hipStream_t stream) {
    (void)in_sizes; (void)n_in; (void)d_ws; (void)ws_size; (void)out_size;
    const float* inp = (const float*)d_in[0];   // (4096, 32, 64) f32
    const float* W   = (const float*)d_in[1];   // (496, 64, 64) f32
    float* out       = (float*)d_out;           // (4096, 496, 64) f32

    dim3 grid(B_TOT / BT, P_);   // (32, 496)
    dim3 block(256);
    bilinear_wmma_f32_kernel<<<grid, block, 0, stream>>>(inp, W, out);
}